// MatchingNetwork_39539468927059
// MI455X (gfx1250) — compile-verified
//
#include <hip/hip_runtime.h>
#include <hip/hip_bf16.h>
#include <math.h>

// ---------------- problem constants ----------------
#define NSAMP 16404
#define LIN   300
#define L1    298      // conv1 out length (pad 1, k5)
#define LP1   59       // pool1 out
#define L2    57       // conv2 out length
#define LP2   11       // pool2 out
#define C1    32
#define VEC   352
#define HID   352
#define G4    1408     // 4*HID
#define NSUP  20
#define NQ    16384    // NSAMP - NSUP

#define STAT_BLOCKS 2048

typedef float v2f __attribute__((ext_vector_type(2)));
typedef float v8f __attribute__((ext_vector_type(8)));

__device__ __forceinline__ float sigm(float x) { return 1.0f / (1.0f + expf(-x)); }

// ================= conv1 BN1 statistics (pass 1) =================
// partial1: STAT_BLOCKS x 64 (sum[32], sumsq[32]) -- deterministic per block
__global__ __launch_bounds__(256)
void bn1_stats_kernel(const float* __restrict__ x,
                      const float* __restrict__ w1, const float* __restrict__ b1,
                      float* __restrict__ partial1) {
  const int t = threadIdx.x;
  const int c = t & 31;
  float w[5];
#pragma unroll
  for (int k = 0; k < 5; ++k) w[k] = w1[c * 5 + k];
  const float bias = b1[c];
  float s = 0.f, sq = 0.f;
  const long total = (long)NSAMP * L1;
  for (long p = (long)blockIdx.x * 8 + (t >> 5); p < total; p += (long)gridDim.x * 8) {
    int n = (int)(p / L1), l = (int)(p % L1);
    const float* xr = x + (size_t)n * LIN;
    float y = bias;
#pragma unroll
    for (int k = 0; k < 5; ++k) {
      int xi = l + k - 1;
      float xv = (xi >= 0 && xi < LIN) ? xr[xi] : 0.f;
      y = fmaf(w[k], xv, y);
    }
    s += y; sq += y * y;
  }
  __shared__ float rs[256], rq[256];
  rs[t] = s; rq[t] = sq;
  __syncthreads();
  if (t < 32) {
    float as = 0.f, aq = 0.f;
    for (int i = t; i < 256; i += 32) { as += rs[i]; aq += rq[i]; }  // fixed order
    partial1[(size_t)blockIdx.x * 64 + c]      = as;
    partial1[(size_t)blockIdx.x * 64 + 32 + c] = aq;
  }
}

// ================= deterministic partial reduction =================
__global__ void reduce_stats_kernel(const float* __restrict__ partials, int nblk,
                                    float* __restrict__ stats) {
  int t = threadIdx.x;  // 64 threads
  float s = 0.f;
  for (int b = 0; b < nblk; ++b) s += partials[(size_t)b * 64 + t];
  stats[t] = s;
}

// ================= BN finalize: scale/shift from sum/sumsq =================
__global__ void bn_finalize_kernel(const float* __restrict__ stats, float count,
                                   const float* __restrict__ g, const float* __restrict__ b,
                                   float* __restrict__ ss) {
  int c = threadIdx.x;  // 32
  float mean = stats[c] / count;
  float var  = stats[32 + c] / count - mean * mean;
  float sc   = g[c] * rsqrtf(var + 1e-5f);
  ss[c]      = sc;
  ss[32 + c] = b[c] - mean * sc;
}

// ====== fused conv1->bn1->relu->pool -> conv2 ; mode 0: bn2 stats, mode 1: z ======
__global__ __launch_bounds__(256)
void conv_pipeline_kernel(const float* __restrict__ x,
                          const float* __restrict__ w1, const float* __restrict__ b1,
                          const float* __restrict__ ss1,   // scale[32], shift[32]
                          const float* __restrict__ w2, const float* __restrict__ b2,
                          int mode,
                          const float* __restrict__ ss2,   // mode 1 only
                          float* __restrict__ partial2,    // mode 0: NSAMP x 64
                          float* __restrict__ z) {         // mode 1: NSAMP x 352
  __shared__ float xs[LIN];
  __shared__ float h1[C1][LP1];
  __shared__ float a2[C1][L2];
  const int n = blockIdx.x;
  const int t = threadIdx.x;
  for (int i = t; i < LIN; i += 256) xs[i] = x[(size_t)n * LIN + i];
  __syncthreads();
  // h1 = pool5(relu(bn1(conv1)))
  for (int idx = t; idx < C1 * LP1; idx += 256) {
    int c = idx / LP1, lp = idx - c * LP1;
    float w[5];
#pragma unroll
    for (int k = 0; k < 5; ++k) w[k] = w1[c * 5 + k];
    float bias = b1[c], sc = ss1[c], sh = ss1[32 + c];
    float m = 0.f;  // relu output >= 0
#pragma unroll
    for (int j = 0; j < 5; ++j) {
      int l = lp * 5 + j;  // <= 294 < 298
      float y = bias;
#pragma unroll
      for (int k = 0; k < 5; ++k) {
        int xi = l + k - 1;
        float xv = (xi >= 0 && xi < LIN) ? xs[xi] : 0.f;
        y = fmaf(w[k], xv, y);
      }
      y = fmaxf(fmaf(y, sc, sh), 0.f);
      m = fmaxf(m, y);
    }
    h1[c][lp] = m;
  }
  __syncthreads();
  // conv2
  for (int idx = t; idx < C1 * L2; idx += 256) {
    int co = idx / L2, l2 = idx - co * L2;
    float y = b2[co];
    for (int ci = 0; ci < C1; ++ci) {
      const float* wr = w2 + ((size_t)co * C1 + ci) * 5;
#pragma unroll
      for (int k = 0; k < 5; ++k) {
        int hl = l2 + k - 1;
        float hv = (hl >= 0 && hl < LP1) ? h1[ci][hl] : 0.f;
        y = fmaf(wr[k], hv, y);
      }
    }
    if (mode) y = fmaxf(fmaf(y, ss2[co], ss2[32 + co]), 0.f);
    a2[co][l2] = y;
  }
  __syncthreads();
  if (mode == 0) {
    if (t < C1) {
      float s = 0.f, sq = 0.f;
      for (int l2 = 0; l2 < L2; ++l2) { float v = a2[t][l2]; s += v; sq += v * v; }
      partial2[(size_t)n * 64 + t]      = s;
      partial2[(size_t)n * 64 + 32 + t] = sq;
    }
  } else {
    for (int idx = t; idx < C1 * LP2; idx += 256) {
      int c = idx / LP2, lp = idx - c * LP2;
      float m = 0.f;
#pragma unroll
      for (int j = 0; j < 5; ++j) m = fmaxf(m, a2[c][lp * 5 + j]);  // <= 54 < 57
      z[(size_t)n * VEC + c * LP2 + lp] = m;
    }
  }
}

// ===== prototypes + shared (query-independent) LSTM chain: 1 block, 352 threads =====
__global__ __launch_bounds__(352)
void lstm_shared_kernel(const float* __restrict__ z,
                        const float* __restrict__ wih_f, const float* __restrict__ whh_f,
                        const float* __restrict__ bih_f, const float* __restrict__ bhh_f,
                        const float* __restrict__ wih_r,
                        const float* __restrict__ bih_r, const float* __restrict__ bhh_r,
                        float* __restrict__ hf0, float* __restrict__ hf1,
                        float* __restrict__ c2f, float* __restrict__ shF,
                        float* __restrict__ shx1, float* __restrict__ shx0) {
  __shared__ float s0[VEC], s1[VEC], h[HID], c[HID];
  const int j = threadIdx.x;  // 0..351
  float a0 = 0.f, a1 = 0.f;
  for (int k = 0; k < 10; ++k) {
    a0 += z[(size_t)k * VEC + j];
    a1 += z[(size_t)(10 + k) * VEC + j];
  }
  s0[j] = a0 * 0.1f; s1[j] = a1 * 0.1f;
  __syncthreads();
  // ---- forward t=0 (x=s0, h=c=0) ----
  float g[4];
  for (int gi = 0; gi < 4; ++gi) {
    int row = j + gi * HID;
    float acc = bih_f[row] + bhh_f[row];
    const float* wr = wih_f + (size_t)row * VEC;
    for (int v = 0; v < VEC; ++v) acc = fmaf(wr[v], s0[v], acc);
    g[gi] = acc;
  }
  float cc = sigm(g[0]) * tanhf(g[2]);          // f*0 + i*tanh(g)
  float hh = sigm(g[3]) * tanhf(cc);
  h[j] = hh; c[j] = cc; hf0[j] = hh;
  __syncthreads();
  // ---- forward t=1 (x=s1) ----
  for (int gi = 0; gi < 4; ++gi) {
    int row = j + gi * HID;
    float acc = bih_f[row] + bhh_f[row];
    const float* wr1 = wih_f + (size_t)row * VEC;
    const float* wr2 = whh_f + (size_t)row * VEC;
    for (int v = 0; v < VEC; ++v) acc = fmaf(wr1[v], s1[v], fmaf(wr2[v], h[v], acc));
    g[gi] = acc;
  }
  __syncthreads();  // all reads of h done before overwriting
  cc = sigm(g[1]) * c[j] + sigm(g[0]) * tanhf(g[2]);
  hh = sigm(g[3]) * tanhf(cc);
  h[j] = hh; c[j] = cc; hf1[j] = hh; c2f[j] = cc;
  __syncthreads();
  // ---- shared t=2 parts: shF = Whh_f@h2 + b ; shx = Wih_r@s + b ----
  for (int gi = 0; gi < 4; ++gi) {
    int row = j + gi * HID;
    float accF = bih_f[row] + bhh_f[row];
    const float* wr = whh_f + (size_t)row * VEC;
    for (int v = 0; v < VEC; ++v) accF = fmaf(wr[v], h[v], accF);
    shF[row] = accF;
    float br = bih_r[row] + bhh_r[row];
    const float* wrr = wih_r + (size_t)row * VEC;
    float r1 = br, r0 = br;
    for (int v = 0; v < VEC; ++v) {
      r1 = fmaf(wrr[v], s1[v], r1);
      r0 = fmaf(wrr[v], s0[v], r0);
    }
    shx1[row] = r1; shx0[row] = r0;
  }
}

// ================= WMMA f32 GEMM: C[M,N] = A[M,K] @ B[N,K]^T =================
// M=16384, N=1408, K=352. Block tile 128x64x32, 8 waves, wave tile 32x32
// (2x2 grid of 16x16 accumulators -> every A/B fragment feeds 2 WMMAs).
#define BM 128
#define BN 64
#define BK 32
__global__ __launch_bounds__(256)
void gemm_wmma_f32(const float* __restrict__ A, const float* __restrict__ B,
                   float* __restrict__ C, int M, int N, int K) {
  __shared__ float As[BM][BK + 1];
  __shared__ float Bs[BN][BK + 1];
  const int t     = threadIdx.x;
  const int lane  = t & 31;
  const int wave  = t >> 5;
  const int mOff  = (wave & 3) << 5;   // 4 M-bands of 32
  const int nOff  = (wave >> 2) << 5;  // 2 N-bands of 32
  const int idx16 = lane & 15;
  const int half  = lane >> 4;
  const int mBlock = blockIdx.x * BM;
  const int nBlock = blockIdx.y * BN;
  v8f acc00 = {0, 0, 0, 0, 0, 0, 0, 0};
  v8f acc01 = {0, 0, 0, 0, 0, 0, 0, 0};
  v8f acc10 = {0, 0, 0, 0, 0, 0, 0, 0};
  v8f acc11 = {0, 0, 0, 0, 0, 0, 0, 0};
  for (int k0 = 0; k0 < K; k0 += BK) {
    // stage A tile: 128x32 (16 elements/thread), B tile: 64x32 (8 elements/thread)
#pragma unroll
    for (int i = 0; i < 16; ++i) {
      int e = t + 256 * i;
      int r = e >> 5, cidx = e & 31;
      As[r][cidx] = A[(size_t)(mBlock + r) * K + k0 + cidx];
    }
#pragma unroll
    for (int i = 0; i < 8; ++i) {
      int e = t + 256 * i;
      int r = e >> 5, cidx = e & 31;
      Bs[r][cidx] = B[(size_t)(nBlock + r) * K + k0 + cidx];
    }
    __syncthreads();
#pragma unroll
    for (int kk = 0; kk < BK; kk += 4) {
      // frag layout (ISA 16x4 f32): lane l -> row idx16, k = kk + reg + 2*half
      v2f a0, a1, b0, b1;
      a0.x = As[mOff + idx16][kk + 2 * half + 0];
      a0.y = As[mOff + idx16][kk + 2 * half + 1];
      a1.x = As[mOff + 16 + idx16][kk + 2 * half + 0];
      a1.y = As[mOff + 16 + idx16][kk + 2 * half + 1];
      b0.x = Bs[nOff + idx16][kk + 2 * half + 0];
      b0.y = Bs[nOff + idx16][kk + 2 * half + 1];
      b1.x = Bs[nOff + 16 + idx16][kk + 2 * half + 0];
      b1.y = Bs[nOff + 16 + idx16][kk + 2 * half + 1];
      acc00 = __builtin_amdgcn_wmma_f32_16x16x4_f32(false, a0, false, b0, (short)0, acc00, false, false);
      acc01 = __builtin_amdgcn_wmma_f32_16x16x4_f32(false, a0, false, b1, (short)0, acc01, false, false);
      acc10 = __builtin_amdgcn_wmma_f32_16x16x4_f32(false, a1, false, b0, (short)0, acc10, false, false);
      acc11 = __builtin_amdgcn_wmma_f32_16x16x4_f32(false, a1, false, b1, (short)0, acc11, false, false);
    }
    __syncthreads();
  }
  // C/D layout: VGPR r -> row r + 8*half, col = idx16
#pragma unroll
  for (int r = 0; r < 8; ++r) {
    int m0 = mBlock + mOff + r + 8 * half;
    int m1 = m0 + 16;
    int n0 = nBlock + nOff + idx16;
    C[(size_t)m0 * N + n0]      = acc00[r];
    C[(size_t)m0 * N + n0 + 16] = acc01[r];
    C[(size_t)m1 * N + n0]      = acc10[r];
    C[(size_t)m1 * N + n0 + 16] = acc11[r];
  }
}

// ================= elementwise LSTM cells =================
__global__ void ew_fwd_kernel(const float* __restrict__ gates, const float* __restrict__ shF,
                              const float* __restrict__ c2f, float* __restrict__ hf2) {
  long idx = (long)blockIdx.x * 256 + threadIdx.x;
  if (idx >= (long)NQ * HID) return;
  int q = (int)(idx / HID), j = (int)(idx - (long)q * HID);
  const float* gq = gates + (size_t)q * G4;
  float gi = gq[j] + shF[j];
  float gf = gq[j + HID] + shF[j + HID];
  float gg = gq[j + 2 * HID] + shF[j + 2 * HID];
  float go = gq[j + 3 * HID] + shF[j + 3 * HID];
  float cc = sigm(gf) * c2f[j] + sigm(gi) * tanhf(gg);
  hf2[idx] = sigm(go) * tanhf(cc);
}

__global__ void ew_b0_kernel(const float* __restrict__ gates,
                             const float* __restrict__ bih, const float* __restrict__ bhh,
                             float* __restrict__ hB1, float* __restrict__ cB) {
  long idx = (long)blockIdx.x * 256 + threadIdx.x;
  if (idx >= (long)NQ * HID) return;
  int q = (int)(idx / HID), j = (int)(idx - (long)q * HID);
  const float* gq = gates + (size_t)q * G4;
  float gi = gq[j] + bih[j] + bhh[j];
  float gg = gq[j + 2 * HID] + bih[j + 2 * HID] + bhh[j + 2 * HID];
  float go = gq[j + 3 * HID] + bih[j + 3 * HID] + bhh[j + 3 * HID];
  float cc = sigm(gi) * tanhf(gg);   // c_prev = 0
  cB[idx]  = cc;
  hB1[idx] = sigm(go) * tanhf(cc);
}

__global__ void ew_bstep_kernel(const float* __restrict__ gates, const float* __restrict__ shx,
                                float* __restrict__ cB, float* __restrict__ hOut) {
  long idx = (long)blockIdx.x * 256 + threadIdx.x;
  if (idx >= (long)NQ * HID) return;
  int q = (int)(idx / HID), j = (int)(idx - (long)q * HID);
  const float* gq = gates + (size_t)q * G4;
  float gi = gq[j] + shx[j];
  float gf = gq[j + HID] + shx[j + HID];
  float gg = gq[j + 2 * HID] + shx[j + 2 * HID];
  float go = gq[j + 3 * HID] + shx[j + 3 * HID];
  float cc = sigm(gf) * cB[idx] + sigm(gi) * tanhf(gg);
  cB[idx]  = cc;
  hOut[idx] = sigm(go) * tanhf(cc);
}

// ================= cosine similarity + softmax (one wave per query) =================
__global__ __launch_bounds__(256)
void out_kernel(const float* __restrict__ hf0, const float* __restrict__ hf1,
                const float* __restrict__ hf2, const float* __restrict__ hB1,
                const float* __restrict__ hB2, const float* __restrict__ hB3,
                float* __restrict__ out) {
  int q = blockIdx.x * 8 + (threadIdx.x >> 5);
  int lane = threadIdx.x & 31;
  if (q >= NQ) return;
  const float* f2 = hf2 + (size_t)q * HID;
  const float* b1 = hB1 + (size_t)q * HID;
  const float* b2 = hB2 + (size_t)q * HID;
  const float* b3 = hB3 + (size_t)q * HID;
  float d00 = 0, d10 = 0, dB3 = 0, dB2 = 0;
  float n2 = 0, nb1 = 0, nb2 = 0, nb3 = 0, n0 = 0, n1 = 0;
  for (int j = lane; j < HID; j += 32) {
    float v2 = f2[j], vb1 = b1[j], vb2 = b2[j], vb3 = b3[j];
    float v0 = hf0[j], v1 = hf1[j];
    d00 += v0 * v2;  d10 += v1 * v2;
    dB3 += vb3 * vb1; dB2 += vb2 * vb1;
    n2 += v2 * v2;  nb1 += vb1 * vb1; nb2 += vb2 * vb2; nb3 += vb3 * vb3;
    n0 += v0 * v0;  n1  += v1 * v1;
  }
#pragma unroll
  for (int off = 16; off > 0; off >>= 1) {
    d00 += __shfl_down(d00, off, 32); d10 += __shfl_down(d10, off, 32);
    dB3 += __shfl_down(dB3, off, 32); dB2 += __shfl_down(dB2, off, 32);
    n2  += __shfl_down(n2,  off, 32); nb1 += __shfl_down(nb1, off, 32);
    nb2 += __shfl_down(nb2, off, 32); nb3 += __shfl_down(nb3, off, 32);
    n0  += __shfl_down(n0,  off, 32); n1  += __shfl_down(n1,  off, 32);
  }
  if (lane == 0) {
    float nq  = fmaxf(sqrtf(n2 + nb1), 1e-8f);
    float ns0 = fmaxf(sqrtf(n0 + nb3), 1e-8f);
    float ns1 = fmaxf(sqrtf(n1 + nb2), 1e-8f);
    float s0 = (d00 + dB3) / (ns0 * nq);
    float s1 = (d10 + dB2) / (ns1 * nq);
    float m = fmaxf(s0, s1);
    float e0 = expf(s0 - m), e1 = expf(s1 - m);
    float inv = 1.f / (e0 + e1);
    out[(size_t)q * 2]     = e0 * inv;
    out[(size_t)q * 2 + 1] = e1 * inv;
  }
}

// ================= host =================
extern "C" void kernel_launch(void* const* d_in, const int* in_sizes, int n_in,
                              void* d_out, int out_size, void* d_ws, size_t ws_size,
                              hipStream_t stream) {
  (void)in_sizes; (void)n_in; (void)out_size; (void)ws_size;
  const float* x       = (const float*)d_in[0];
  const float* conv1_w = (const float*)d_in[1];
  const float* conv1_b = (const float*)d_in[2];
  const float* bn1_g   = (const float*)d_in[3];
  const float* bn1_b   = (const float*)d_in[4];
  const float* conv2_w = (const float*)d_in[5];
  const float* conv2_b = (const float*)d_in[6];
  const float* bn2_g   = (const float*)d_in[7];
  const float* bn2_b   = (const float*)d_in[8];
  const float* w_ih_f  = (const float*)d_in[9];
  const float* w_hh_f  = (const float*)d_in[10];
  const float* b_ih_f  = (const float*)d_in[11];
  const float* b_hh_f  = (const float*)d_in[12];
  const float* w_ih_r  = (const float*)d_in[13];
  const float* w_hh_r  = (const float*)d_in[14];
  const float* b_ih_r  = (const float*)d_in[15];
  const float* b_hh_r  = (const float*)d_in[16];

  float* ws = (float*)d_ws;
  size_t o = 0;
  float* stats1   = ws + o; o += 64;
  float* ss1      = ws + o; o += 64;
  float* stats2   = ws + o; o += 64;
  float* ss2      = ws + o; o += 64;
  float* partial1 = ws + o; o += (size_t)STAT_BLOCKS * 64;
  float* partial2 = ws + o; o += (size_t)NSAMP * 64;
  float* z        = ws + o; o += (size_t)NSAMP * VEC;
  float* hf0      = ws + o; o += HID;
  float* hf1      = ws + o; o += HID;
  float* c2f      = ws + o; o += HID;
  float* shF      = ws + o; o += G4;
  float* shx1     = ws + o; o += G4;
  float* shx0     = ws + o; o += G4;
  float* gates    = ws + o; o += (size_t)NQ * G4;
  float* hf2      = ws + o; o += (size_t)NQ * HID;
  float* hB1      = ws + o; o += (size_t)NQ * HID;
  float* hB2      = ws + o; o += (size_t)NQ * HID;
  float* hB3      = ws + o; o += (size_t)NQ * HID;
  float* cB       = ws + o; o += (size_t)NQ * HID;

  // ---- conv stack with batch-norm (two passes per layer) ----
  bn1_stats_kernel<<<STAT_BLOCKS, 256, 0, stream>>>(x, conv1_w, conv1_b, partial1);
  reduce_stats_kernel<<<1, 64, 0, stream>>>(partial1, STAT_BLOCKS, stats1);
  bn_finalize_kernel<<<1, 32, 0, stream>>>(stats1, (float)((long)NSAMP * L1), bn1_g, bn1_b, ss1);
  conv_pipeline_kernel<<<NSAMP, 256, 0, stream>>>(x, conv1_w, conv1_b, ss1, conv2_w, conv2_b,
                                                  0, nullptr, partial2, nullptr);
  reduce_stats_kernel<<<1, 64, 0, stream>>>(partial2, NSAMP, stats2);
  bn_finalize_kernel<<<1, 32, 0, stream>>>(stats2, (float)((long)NSAMP * L2), bn2_g, bn2_b, ss2);
  conv_pipeline_kernel<<<NSAMP, 256, 0, stream>>>(x, conv1_w, conv1_b, ss1, conv2_w, conv2_b,
                                                  1, ss2, nullptr, z);

  // ---- query-independent LSTM chain ----
  lstm_shared_kernel<<<1, 352, 0, stream>>>(z, w_ih_f, w_hh_f, b_ih_f, b_hh_f,
                                            w_ih_r, b_ih_r, b_hh_r,
                                            hf0, hf1, c2f, shF, shx1, shx0);

  const float* zq = z + (size_t)NSUP * VEC;
  dim3 ggrid(NQ / BM, G4 / BN);
  long tot = (long)NQ * HID;
  int ewb = (int)((tot + 255) / 256);

  // forward t=2: gates = query @ Wih_f^T ; cell with shared (shF, c2f)
  gemm_wmma_f32<<<ggrid, 256, 0, stream>>>(zq, w_ih_f, gates, NQ, G4, VEC);
  ew_fwd_kernel<<<ewb, 256, 0, stream>>>(gates, shF, c2f, hf2);
  // backward step 0: gates = query @ Wih_r^T
  gemm_wmma_f32<<<ggrid, 256, 0, stream>>>(zq, w_ih_r, gates, NQ, G4, VEC);
  ew_b0_kernel<<<ewb, 256, 0, stream>>>(gates, b_ih_r, b_hh_r, hB1, cB);
  // backward step 1: gates = hB1 @ Whh_r^T ; x-part = shx1
  gemm_wmma_f32<<<ggrid, 256, 0, stream>>>(hB1, w_hh_r, gates, NQ, G4, VEC);
  ew_bstep_kernel<<<ewb, 256, 0, stream>>>(gates, shx1, cB, hB2);
  // backward step 2: gates = hB2 @ Whh_r^T ; x-part = shx0
  gemm_wmma_f32<<<ggrid, 256, 0, stream>>>(hB2, w_hh_r, gates, NQ, G4, VEC);
  ew_bstep_kernel<<<ewb, 256, 0, stream>>>(gates, shx0, cB, hB3);

  // ---- cosine similarity + softmax ----
  out_kernel<<<(NQ + 7) / 8, 256, 0, stream>>>(hf0, hf1, hf2, hB1, hB2, hB3, (float*)d_out);
}